// Permutator_5970004542088
// MI455X (gfx1250) — compile-verified
//
#include <hip/hip_runtime.h>
#include <cstdint>

// Problem constants from the reference.
#define DEPTH 4096   // WAVE_DEPTH (columns of x, size of perm)
#define NROWS 8192   // N_ROWS

// ---------------------------------------------------------------------------
// Kernel A: extract inverse permutation from the one-hot matrix.
//   perm[k][c] == 1  =>  inv[c] = k   (out[r,c] = x[r, inv[c]])
// Pure streaming read of 64 MB with b128 loads; writes 16 KB of indices.
// ---------------------------------------------------------------------------
__global__ __launch_bounds__(256) void build_inv_kernel(
    const float* __restrict__ perm, int* __restrict__ inv) {
  unsigned gid = blockIdx.x * 256u + threadIdx.x;      // 0 .. 4M-1 (float4 units)
  const float4 v = reinterpret_cast<const float4*>(perm)[gid];
  unsigned e = gid * 4u;                               // flat element index
  unsigned k = e >> 12;                                // row    (e / 4096)
  unsigned c = e & 4095u;                              // column (e % 4096)
  if (v.x != 0.0f) inv[c + 0] = (int)k;
  if (v.y != 0.0f) inv[c + 1] = (int)k;
  if (v.z != 0.0f) inv[c + 2] = (int)k;
  if (v.w != 0.0f) inv[c + 3] = (int)k;
}

// ---------------------------------------------------------------------------
// Kernel B: one block per row of x.
//   Phase 1: stage the 16 KB row into LDS with CDNA5 async global->LDS DMA
//            (global_load_async_to_lds_b128, ASYNCcnt).  HBM reads stay
//            perfectly streaming / 128-bit.
//   Phase 2: random permutation happens inside LDS (ds_load_b32 gathers),
//            then coalesced 128-bit stores of the output row.
// ---------------------------------------------------------------------------
__global__ __launch_bounds__(1024) void permute_rows_kernel(
    const float* __restrict__ x, const int* __restrict__ inv,
    float* __restrict__ out) {
  __shared__ float smem[DEPTH];                        // 16 KB row buffer
  const unsigned t = threadIdx.x;                      // 0..1023
  const unsigned r = blockIdx.x;                       // row index

  // --- Phase 1: async copy x[r, t*4 .. t*4+3] (b128 per lane) into LDS ---
  {
    const float* src = x + (size_t)r * DEPTH + (size_t)t * 4u;
    uint64_t gaddr = (uint64_t)(uintptr_t)src;
    // Generic pointer to LDS: low 32 bits are the LDS byte offset.
    uint32_t laddr = (uint32_t)(uintptr_t)(&smem[t * 4u]);
    asm volatile("global_load_async_to_lds_b128 %0, %1, off"
                 :
                 : "v"(laddr), "v"(gaddr)
                 : "memory");
    asm volatile("s_wait_asynccnt 0" ::: "memory");
  }
  __syncthreads();

  // --- Phase 2: gather 4 permuted columns from LDS, b128 store ---
  const int4 iv = reinterpret_cast<const int4*>(inv)[t];  // coalesced, L2-hot
  float4 o;
  o.x = smem[iv.x];
  o.y = smem[iv.y];
  o.z = smem[iv.z];
  o.w = smem[iv.w];
  reinterpret_cast<float4*>(out + (size_t)r * DEPTH)[t] = o;
}

// ---------------------------------------------------------------------------
// Launcher.  d_in[0] = x (8192x4096 f32), d_in[1] = perm (4096x4096 f32).
// d_ws holds the 16 KB inverse-index vector (recomputed every call; launcher
// is stateless and deterministic).
// ---------------------------------------------------------------------------
extern "C" void kernel_launch(void* const* d_in, const int* in_sizes, int n_in,
                              void* d_out, int out_size, void* d_ws, size_t ws_size,
                              hipStream_t stream) {
  const float* x    = (const float*)d_in[0];
  const float* perm = (const float*)d_in[1];
  float* out        = (float*)d_out;
  int*   inv        = (int*)d_ws;     // needs DEPTH * 4 = 16 KB of scratch

  // Kernel A: 4096*4096 elements, 4 per thread, 256 threads/block.
  const unsigned nblkA = (DEPTH * (unsigned)DEPTH) / 4u / 256u;  // 16384
  build_inv_kernel<<<dim3(nblkA), dim3(256), 0, stream>>>(perm, inv);

  // Kernel B: one block per row.
  permute_rows_kernel<<<dim3(NROWS), dim3(1024), 0, stream>>>(x, inv, out);
}